// PennyLaneModule_80144089744002
// MI455X (gfx1250) — compile-verified
//
#include <hip/hip_runtime.h>
#include <math.h>

typedef __attribute__((ext_vector_type(2))) float v2f;
typedef __attribute__((ext_vector_type(8))) float v8f;

#define NQ    16
#define BATCH 256

static __device__ __forceinline__ float2 cmul(float2 a, float2 b) {
  return make_float2(a.x * b.x - a.y * b.y, a.x * b.y + a.y * b.x);
}
static __device__ __forceinline__ v2f mk2(float a, float b) {
  v2f r; r.x = a; r.y = b; return r;
}

// D(16x16,f32) = A(16x4,f32) x B(4x16,f32) + C   -- CDNA5 V_WMMA_F32_16X16X4_F32
static __device__ __forceinline__ v8f wmma4(v2f a, v2f b, v8f c) {
#if defined(__gfx1250__) && __has_builtin(__builtin_amdgcn_wmma_f32_16x16x4_f32)
  // 8-arg pattern (probe family): (neg_a, A, neg_b, B, c_mod, C, reuse_a, reuse_b)
  // NEG[1:0] must be 0 for f32 per ISA; negation done in VALU on fragments.
  return __builtin_amdgcn_wmma_f32_16x16x4_f32(false, a, false, b, (short)0, c,
                                               false, false);
#else
  c[0] += a.x * b.x;  // host-pass placeholder; never executed on device
  return c;
#endif
}

// ---------------------------------------------------------------------------
// Kernel 1: build per-batch fused nibble matrices U_n (16x16 complex), n=0..3
// U_n acts on index bits [15-4n : 12-4n]; U_n = G_{4n} (x) ... (x) G_{4n+3}
// G_w = RX(p[w,1]) * RZ(p[w,0]) * RY(x[b,w]*pi)
// ---------------------------------------------------------------------------
__global__ void k_build(const float* __restrict__ x, const float* __restrict__ p,
                        float2* __restrict__ U) {
  __shared__ float2 g[16][2][2];
  const int b = blockIdx.x, t = threadIdx.x;
  if (t < 16) {
    const int w = t;
    const float th = x[b * NQ + w] * 3.14159265358979323846f * 0.5f;
    const float c = cosf(th), s = sinf(th);
    const float ph = p[w * 2 + 0] * 0.5f;
    const float er = cosf(ph), ei = -sinf(ph);
    // M = RZ * RY  (RZ = diag(e, conj(e)))
    const float2 M00 = make_float2(er * c, ei * c);
    const float2 M01 = make_float2(-er * s, -ei * s);
    const float2 M10 = make_float2(er * s, -ei * s);
    const float2 M11 = make_float2(er * c, -ei * c);
    const float lm = p[w * 2 + 1] * 0.5f;
    const float c2 = cosf(lm);
    const float2 f = make_float2(0.f, -sinf(lm));  // -i*sin
    float2 t00 = cmul(f, M10), t01 = cmul(f, M11);
    float2 t10 = cmul(f, M00), t11 = cmul(f, M01);
    g[w][0][0] = make_float2(c2 * M00.x + t00.x, c2 * M00.y + t00.y);
    g[w][0][1] = make_float2(c2 * M01.x + t01.x, c2 * M01.y + t01.y);
    g[w][1][0] = make_float2(t10.x + c2 * M10.x, t10.y + c2 * M10.y);
    g[w][1][1] = make_float2(t11.x + c2 * M11.x, t11.y + c2 * M11.y);
  }
  __syncthreads();
  const int i = t >> 4, j = t & 15;
  for (int n = 0; n < 4; ++n) {
    float2 pr = g[4 * n + 0][(i >> 3) & 1][(j >> 3) & 1];
    pr = cmul(pr, g[4 * n + 1][(i >> 2) & 1][(j >> 2) & 1]);
    pr = cmul(pr, g[4 * n + 2][(i >> 1) & 1][(j >> 1) & 1]);
    pr = cmul(pr, g[4 * n + 3][i & 1][j & 1]);
    U[(((b << 2) + n) << 8) + t] = pr;
  }
}

// ---------------------------------------------------------------------------
// Kernel 2: layer-1 state = G|0..0>  (factorized: product of U_n first columns)
// ---------------------------------------------------------------------------
__global__ void k_init(const float2* __restrict__ U, float2* __restrict__ dst) {
  __shared__ float2 uc[4][16];
  const int b = blockIdx.x >> 8, c = blockIdx.x & 255, t = threadIdx.x;
  if (t < 64) {
    const int n = t >> 4, k = t & 15;
    uc[n][k] = U[(((b << 2) + n) << 8) + (k << 4)];  // U_n[k][0]
  }
  __syncthreads();
  const int idx = (c << 8) | t;
  float2 a = cmul(cmul(uc[0][(idx >> 12) & 15], uc[1][(idx >> 8) & 15]),
                  cmul(uc[2][(idx >> 4) & 15], uc[3][idx & 15]));
  dst[(b << 16) + idx] = a;
}

// ---------------------------------------------------------------------------
// Kernel 3: low-nibble gates (wires 8-15 -> index bits 7:0) with the previous
// layer's Gray (CNOT-ladder) permutation fused into the LDS load.
// One wave32 per 256-amplitude chunk; 2 complex 16x16x16 GEMMs via WMMA.
// ---------------------------------------------------------------------------
__global__ void __launch_bounds__(256) k_low(const float2* __restrict__ src,
                                             float2* __restrict__ dst,
                                             const float2* __restrict__ U) {
  __shared__ float sRe[8][272];
  __shared__ float sIm[8][272];
  const int b = blockIdx.x >> 5;
  const int grp = blockIdx.x & 31;
  const int t = threadIdx.x;
  const int wave = t >> 5, lane = t & 31;
  const int h = (grp << 3) | wave;      // output chunk (bits 15:8)
  const int n = lane & 15, hh = lane >> 4;
  const float2* U3 = U + (((b << 2) + 3) << 8);  // wires 12-15 -> bits 3:0
  const float2* U2 = U + (((b << 2) + 2) << 8);  // wires 8-11  -> bits 7:4

  // B-fragments of W3^T  (B[K=j0,N=i0] = U3[i0][j0]) and A-fragments of W2
  v2f b3r[4], b3i[4], a2r[4], a2i[4], a2in[4];
#pragma unroll
  for (int k = 0; k < 4; ++k) {
    const int kk = 4 * k + 2 * hh;
    float2 w0 = U3[n * 16 + kk], w1 = U3[n * 16 + kk + 1];
    b3r[k] = mk2(w0.x, w1.x); b3i[k] = mk2(w0.y, w1.y);
    float2 a0 = U2[n * 16 + kk], a1 = U2[n * 16 + kk + 1];
    a2r[k] = mk2(a0.x, a1.x); a2i[k] = mk2(a0.y, a1.y);
    a2in[k] = -a2i[k];
  }

  // source chunk under Gray permutation: high byte = gray8(h)
  const int hs = h ^ (h >> 1);
  const float2* sp = src + (b << 16) + (hs << 8);
  for (int m = lane; m < 256; m += 32) {
    float2 v = sp[m];
    const int pos = (m & 15) + (m >> 4) * 17;
    sRe[wave][pos] = v.x; sIm[wave][pos] = v.y;
  }
  __syncthreads();

  // Phase A: A = T (rows j1, cols j0; intra-chunk Gray perm on read), B = W3^T
  const int flip = (h & 1) << 7;
  v2f tr[4], ti[4], tin[4];
#pragma unroll
  for (int k = 0; k < 4; ++k) {
    const int m0 = n * 16 + 4 * k + 2 * hh;
    const int m1 = m0 + 1;
    const int s0 = (m0 ^ (m0 >> 1)) ^ flip;
    const int s1 = (m1 ^ (m1 >> 1)) ^ flip;
    const int p0 = (s0 & 15) + (s0 >> 4) * 17;
    const int p1 = (s1 & 15) + (s1 >> 4) * 17;
    tr[k] = mk2(sRe[wave][p0], sRe[wave][p1]);
    ti[k] = mk2(sIm[wave][p0], sIm[wave][p1]);
    tin[k] = -ti[k];
  }
  __syncthreads();  // all fragment reads done before LDS is overwritten

  v8f cr = {}, ci = {};
#pragma unroll
  for (int k = 0; k < 4; ++k) {
    cr = wmma4(tr[k], b3r[k], cr);
    cr = wmma4(tin[k], b3i[k], cr);
    ci = wmma4(tr[k], b3i[k], ci);
    ci = wmma4(ti[k], b3r[k], ci);
  }
  // store intermediate A' = T * W3^T back to LDS ([j1][i0])
#pragma unroll
  for (int v = 0; v < 8; ++v) {
    const int M = v + 8 * hh;
    sRe[wave][n + M * 17] = cr[v];
    sIm[wave][n + M * 17] = ci[v];
  }
  __syncthreads();

  // Phase B: A = W2, B = A'
  v8f dr = {}, di = {};
#pragma unroll
  for (int k = 0; k < 4; ++k) {
    const int r0 = 4 * k + 2 * hh;
    v2f br = mk2(sRe[wave][n + r0 * 17], sRe[wave][n + (r0 + 1) * 17]);
    v2f bi = mk2(sIm[wave][n + r0 * 17], sIm[wave][n + (r0 + 1) * 17]);
    dr = wmma4(a2r[k], br, dr);
    dr = wmma4(a2in[k], bi, dr);
    di = wmma4(a2r[k], bi, di);
    di = wmma4(a2i[k], br, di);
  }
  float2* op = dst + (b << 16) + (h << 8);
#pragma unroll
  for (int v = 0; v < 8; ++v) {
    const int M = v + 8 * hh;
    op[M * 16 + n] = make_float2(dr[v], di[v]);
  }
}

// ---------------------------------------------------------------------------
// Kernel 4: high-nibble gates (wires 0-7 -> index bits 15:8). In-place tiles:
// one workgroup = 16 m-columns x all 256 high indices (32 KB tile in LDS).
// ---------------------------------------------------------------------------
__global__ void __launch_bounds__(256) k_high(float2* __restrict__ buf,
                                              const float2* __restrict__ U) {
  __shared__ float tRe[4352];
  __shared__ float tIm[4352];
  const int b = blockIdx.x >> 4;
  const int m0 = (blockIdx.x & 15) << 4;
  const int t = threadIdx.x;
  const int wave = t >> 5, lane = t & 31;
  const int n = lane & 15, hh = lane >> 4;
  const float2* U1 = U + (((b << 2) + 1) << 8);  // wires 4-7 -> bits 11:8
  const float2* U0 = U + (((b << 2) + 0) << 8);  // wires 0-3 -> bits 15:12

  v2f a1r[4], a1i[4], a1in[4], a0r[4], a0i[4], a0in[4];
#pragma unroll
  for (int k = 0; k < 4; ++k) {
    const int kk = 4 * k + 2 * hh;
    float2 w0 = U1[n * 16 + kk], w1 = U1[n * 16 + kk + 1];
    a1r[k] = mk2(w0.x, w1.x); a1i[k] = mk2(w0.y, w1.y); a1in[k] = -a1i[k];
    float2 q0 = U0[n * 16 + kk], q1 = U0[n * 16 + kk + 1];
    a0r[k] = mk2(q0.x, q1.x); a0i[k] = mk2(q0.y, q1.y); a0in[k] = -a0i[k];
  }

  const float2* gp = buf + (b << 16) + m0;
  for (int e = t; e < 4096; e += 256) {
    const int row = e >> 4, col = e & 15;
    float2 v = gp[row * 256 + col];
    tRe[row * 17 + col] = v.x; tIm[row * 17 + col] = v.y;
  }
  __syncthreads();

  // Phase 1: contract lo (bits 11:8) with W1; each wave owns 2 hi-blocks
#pragma unroll
  for (int rep = 0; rep < 2; ++rep) {
    const int hi = wave * 2 + rep;
    v2f br[4], bi[4];
#pragma unroll
    for (int k = 0; k < 4; ++k) {
      const int r0 = hi * 16 + 4 * k + 2 * hh;
      br[k] = mk2(tRe[r0 * 17 + n], tRe[(r0 + 1) * 17 + n]);
      bi[k] = mk2(tIm[r0 * 17 + n], tIm[(r0 + 1) * 17 + n]);
    }
    v8f cr = {}, ci = {};
#pragma unroll
    for (int k = 0; k < 4; ++k) {
      cr = wmma4(a1r[k], br[k], cr);
      cr = wmma4(a1in[k], bi[k], cr);
      ci = wmma4(a1r[k], bi[k], ci);
      ci = wmma4(a1i[k], br[k], ci);
    }
#pragma unroll
    for (int v = 0; v < 8; ++v) {
      const int row = hi * 16 + v + 8 * hh;
      tRe[row * 17 + n] = cr[v]; tIm[row * 17 + n] = ci[v];
    }
  }
  __syncthreads();

  // Phase 2: contract hi (bits 15:12) with W0; write result to global in-place
#pragma unroll
  for (int rep = 0; rep < 2; ++rep) {
    const int lo = wave * 2 + rep;
    v2f br[4], bi[4];
#pragma unroll
    for (int k = 0; k < 4; ++k) {
      const int h0 = 4 * k + 2 * hh;
      br[k] = mk2(tRe[(h0 * 16 + lo) * 17 + n], tRe[((h0 + 1) * 16 + lo) * 17 + n]);
      bi[k] = mk2(tIm[(h0 * 16 + lo) * 17 + n], tIm[((h0 + 1) * 16 + lo) * 17 + n]);
    }
    v8f dr = {}, di = {};
#pragma unroll
    for (int k = 0; k < 4; ++k) {
      dr = wmma4(a0r[k], br[k], dr);
      dr = wmma4(a0in[k], bi[k], dr);
      di = wmma4(a0r[k], bi[k], di);
      di = wmma4(a0i[k], br[k], di);
    }
#pragma unroll
    for (int v = 0; v < 8; ++v) {
      const int M = v + 8 * hh;
      buf[(b << 16) + M * 4096 + lo * 256 + m0 + n] = make_float2(dr[v], di[v]);
    }
  }
}

// ---------------------------------------------------------------------------
// Kernel 5: final CNOT-ladder permutation  dst[o] = src[o ^ (o>>1)]
// (chunk-level gray on high byte + intra-chunk gray via LDS)
// ---------------------------------------------------------------------------
__global__ void k_perm(const float2* __restrict__ src, float2* __restrict__ dst) {
  __shared__ float2 ld[256];
  const int b = blockIdx.x >> 8, c = blockIdx.x & 255, t = threadIdx.x;
  const int cs = c ^ (c >> 1);
  ld[t] = src[(b << 16) + (cs << 8) + t];
  __syncthreads();
  const int s = (t ^ (t >> 1)) ^ ((c & 1) << 7);
  dst[(b << 16) + (c << 8) + t] = ld[s];
}

// ---------------------------------------------------------------------------
extern "C" void kernel_launch(void* const* d_in, const int* in_sizes, int n_in,
                              void* d_out, int out_size, void* d_ws, size_t ws_size,
                              hipStream_t stream) {
  (void)in_sizes; (void)n_in; (void)out_size; (void)ws_size;
  const float* x = (const float*)d_in[0];      // (256,16) f32
  const float* p = (const float*)d_in[1];      // (16,2)  f32
  float2* U  = (float2*)d_ws;                  // 256*4*256 float2 = 2 MB
  float2* X2 = (float2*)d_ws + (size_t)BATCH * 4 * 256;  // 128 MB state scratch
  float2* X1 = (float2*)d_out;                 // 128 MB state (output buffer)

  k_build<<<BATCH, 256, 0, stream>>>(x, p, U);
  k_init<<<BATCH * 256, 256, 0, stream>>>(U, X2);        // X2 = G|0>
  // layer 2: gray-gather(P1) + low gates, then high gates in-place
  k_low <<<BATCH * 32, 256, 0, stream>>>(X2, X1, U);
  k_high<<<BATCH * 16, 256, 0, stream>>>(X1, U);
  // layer 3
  k_low <<<BATCH * 32, 256, 0, stream>>>(X1, X2, U);
  k_high<<<BATCH * 16, 256, 0, stream>>>(X2, U);
  // final CNOT-ladder permutation -> d_out
  k_perm<<<BATCH * 256, 256, 0, stream>>>(X2, X1);
}